// BigraphLightModel_51831665328520
// MI455X (gfx1250) — compile-verified
//
#include <hip/hip_runtime.h>

#define CHUNK 512          // edges staged per block in the SpMM
#define TB 256

typedef float v2f __attribute__((ext_vector_type(2)));
typedef float v8f __attribute__((ext_vector_type(8)));

static __device__ __forceinline__ unsigned lds_off_u32(const void* p) {
  // generic pointers to LDS carry the LDS byte offset in the low 32 bits
  return (unsigned)(unsigned long long)p;
}

// ---------------- elementwise helpers ----------------

__global__ void fillf_kernel(float* __restrict__ p, long n) {
  long i = blockIdx.x * (long)blockDim.x + threadIdx.x;
  if (i < n) p[i] = 0.0f;
}

__global__ void fill4_kernel(float4* __restrict__ p, long n4) {
  long i = blockIdx.x * (long)blockDim.x + threadIdx.x;
  if (i < n4) p[i] = make_float4(0.f, 0.f, 0.f, 0.f);
}

__global__ void copy4_kernel(const float4* __restrict__ in, float4* __restrict__ o, long n4) {
  long i = blockIdx.x * (long)blockDim.x + threadIdx.x;
  if (i < n4) o[i] = in[i];
}

__global__ void scale4_kernel(const float4* __restrict__ in, float4* __restrict__ o,
                              float a, long n4) {
  long i = blockIdx.x * (long)blockDim.x + threadIdx.x;
  if (i < n4) {
    float4 v = in[i];
    o[i] = make_float4(a * v.x, a * v.y, a * v.z, a * v.w);
  }
}

__global__ void copy_scale_kernel(const float4* __restrict__ in, float4* __restrict__ x,
                                  float4* __restrict__ o, float a, long n4) {
  long i = blockIdx.x * (long)blockDim.x + threadIdx.x;
  if (i < n4) {
    float4 v = in[i];
    x[i] = v;
    o[i] = make_float4(a * v.x, a * v.y, a * v.z, a * v.w);
  }
}

__global__ void axpy_kernel(const float* __restrict__ x, float* __restrict__ o,
                            float a, long n) {
  long i = blockIdx.x * (long)blockDim.x + threadIdx.x;
  if (i < n) o[i] += a * x[i];
}

// write h_ii rows into the big table at item_idx
__global__ void item_scatter_kernel(const int* __restrict__ idx,
                                    const float4* __restrict__ h,
                                    float4* __restrict__ x, int n_items) {
  long t = blockIdx.x * (long)blockDim.x + threadIdx.x;
  long i = t >> 4;          // row
  int  c = (int)(t & 15);   // float4 column within D=64
  if (i < n_items) x[(long)idx[i] * 16 + c] = h[i * 16 + c];
}

// ---------------- graph norm ----------------

__global__ void deg_kernel(const int* __restrict__ dst, const float* __restrict__ w,
                           float* __restrict__ deg, int E) {
  int i = blockIdx.x * blockDim.x + threadIdx.x;
  if (i < E) atomicAdd(&deg[dst[i]], w[i]);
}

__global__ void norm_kernel(const int* __restrict__ src, const int* __restrict__ dst,
                            const float* __restrict__ w, const float* __restrict__ deg,
                            float* __restrict__ nrm, int E) {
  int i = blockIdx.x * blockDim.x + threadIdx.x;
  if (i < E) {
    float ds = deg[src[i]], dd = deg[dst[i]];
    float is = ds > 0.f ? rsqrtf(ds) : 0.f;
    float id = dd > 0.f ? rsqrtf(dd) : 0.f;
    nrm[i] = is * w[i] * id;
  }
}

// ---------------- SpMM scatter: y[dst] += nrm * x[src] ----------------
// 16 lanes per edge (float4 per lane), edge stream staged into LDS via the
// CDNA5 async-to-LDS path (ASYNCcnt). D = 64 floats per row.

__global__ __launch_bounds__(TB) void spmm_kernel(
    const int* __restrict__ src, const int* __restrict__ dst,
    const float* __restrict__ nrm, const float* __restrict__ x,
    float* __restrict__ y, int E) {
  __shared__ int   s_src[CHUNK];
  __shared__ int   s_dst[CHUNK];
  __shared__ float s_nrm[CHUNK];

  const int e0 = blockIdx.x * CHUNK;
  int nE = E - e0;
  if (nE > CHUNK) nE = CHUNK;

  if (nE == CHUNK) {
    // full block: async DMA the 3 edge arrays (2 KB each) into LDS
    const int CPA = CHUNK / 4;  // 16B chunks per array
    for (int c = threadIdx.x; c < 3 * CPA; c += blockDim.x) {
      int arr = c / CPA;
      int idx = c - arr * CPA;
      unsigned lds;
      const char* g;
      if (arr == 0)      { lds = lds_off_u32(s_src); g = (const char*)(src + e0); }
      else if (arr == 1) { lds = lds_off_u32(s_dst); g = (const char*)(dst + e0); }
      else               { lds = lds_off_u32(s_nrm); g = (const char*)(nrm + e0); }
      lds += (unsigned)idx * 16u;
      g   += (size_t)idx * 16u;
      asm volatile("global_load_async_to_lds_b128 %0, %1, off"
                   :: "v"(lds), "v"(g) : "memory");
    }
    asm volatile("s_wait_asynccnt 0x0" ::: "memory");
  } else {
    // tail block: plain loads (avoids OOB async reads)
    for (int i = threadIdx.x; i < nE; i += blockDim.x) {
      s_src[i] = src[e0 + i];
      s_dst[i] = dst[e0 + i];
      s_nrm[i] = nrm[e0 + i];
    }
  }
  __syncthreads();

  const int lane = threadIdx.x & 31;
  const int wid  = threadIdx.x >> 5;   // 0..7
  const int half = lane >> 4;          // two edges per wave
  const int l16  = lane & 15;

  for (int e = wid * 2 + half; e < nE; e += 16) {
    const int   s  = s_src[e];
    const int   d  = s_dst[e];
    const float nv = s_nrm[e];
    const float4 v = ((const float4*)(x + (long)s * 64))[l16];
    float* yr = y + (long)d * 64 + l16 * 4;
    atomicAdd(yr + 0, nv * v.x);
    atomicAdd(yr + 1, nv * v.y);
    atomicAdd(yr + 2, nv * v.z);
    atomicAdd(yr + 3, nv * v.w);
  }
}

// ---------------- out += alpha * x via V_WMMA_F32_16X16X4_F32 ----------------
// One wave per 16x16 tile. D = C + sum_j A_j x B_j where B_j[k,n] = a*delta(n,4j+k)
// selects column group j, so D == C + a*X exactly (one nonzero k-term per output).

__global__ __launch_bounds__(TB) void combine_wmma_kernel(
    float* __restrict__ out, const float* __restrict__ x,
    float alpha, int n_row_tiles) {
  const int wave = blockIdx.x * (blockDim.x >> 5) + (threadIdx.x >> 5);
  const int n_tiles = n_row_tiles * 4;          // 4 column groups of 16 (D=64)
  if (wave >= n_tiles) return;                  // wave-uniform exit, EXEC stays full

  const int lane = threadIdx.x & 31;
  const int hi   = lane >> 4;                   // 0: rows M..M+7 / K0,K1 ; 1: +8 / K2,K3
  const int ln   = lane & 15;
  const int m0   = (wave >> 2) << 4;
  const int n0   = (wave & 3) << 4;

  // C/D layout: VGPR p, lanes 0-15 -> (M=m0+p, N=n0+ln); lanes 16-31 -> M=m0+p+8
  v8f c;
  const float* cbase = out + (long)(m0 + 8 * hi) * 64 + n0 + ln;
#pragma unroll
  for (int p = 0; p < 8; ++p) c[p] = cbase[(long)p * 64];

  const float* xrow = x + (long)(m0 + ln) * 64 + n0;  // A: M = ln for both halves
#pragma unroll
  for (int j = 0; j < 4; ++j) {
    const int k2 = 4 * j + 2 * hi;              // A VGPR0/1 hold K = k2, k2+1
    v2f a;
    a.x = xrow[k2];
    a.y = xrow[k2 + 1];
    v2f b;                                       // B VGPR0/1 hold rows K = 2*hi, 2*hi+1
    b.x = (ln == k2)     ? alpha : 0.0f;
    b.y = (ln == k2 + 1) ? alpha : 0.0f;
    c = __builtin_amdgcn_wmma_f32_16x16x4_f32(false, a, false, b, (short)0, c,
                                              false, false);
  }

  float* obase = out + (long)(m0 + 8 * hi) * 64 + n0 + ln;
#pragma unroll
  for (int p = 0; p < 8; ++p) obase[(long)p * 64] = c[p];
}

// ---------------- host orchestration ----------------

static inline int cdiv(long a, long b) { return (int)((a + b - 1) / b); }

static void run_combine(float* out, const float* x, float alpha, int N, hipStream_t stream) {
  const int rt = N / 16;
  if (rt > 0) {
    const int tiles = rt * 4;
    combine_wmma_kernel<<<cdiv(tiles, TB / 32), TB, 0, stream>>>(out, x, alpha, rt);
  }
  const int rem = N - rt * 16;
  if (rem > 0) {
    const long off = (long)rt * 16 * 64;
    const long n = (long)rem * 64;
    axpy_kernel<<<cdiv(n, TB), TB, 0, stream>>>(x + off, out + off, alpha, n);
  }
}

extern "C" void kernel_launch(void* const* d_in, const int* in_sizes, int n_in,
                              void* d_out, int out_size, void* d_ws, size_t ws_size,
                              hipStream_t stream) {
  const float* emb_ii  = (const float*)d_in[0];
  const float* emb_uiu = (const float*)d_in[1];
  const float* w_ii    = (const float*)d_in[2];
  const float* w_uiu   = (const float*)d_in[3];
  const int*   ei_ii   = (const int*)d_in[4];
  const int*   ei_uiu  = (const int*)d_in[5];
  const int*   itemidx = (const int*)d_in[6];
  float* out = (float*)d_out;

  const int D     = 64;
  const int N_ii  = in_sizes[0] / D;
  const int N_uiu = in_sizes[1] / D;
  const int E_ii  = in_sizes[2];
  const int E_uiu = in_sizes[3];
  const int n_items = in_sizes[6];
  const int L = 3;
  const float alpha = 1.0f / (float)(L + 1);

  const int* src_ii = ei_ii;  const int* dst_ii = ei_ii + E_ii;
  const int* src_u  = ei_uiu; const int* dst_u  = ei_uiu + E_uiu;

  // workspace layout (all regions 256B-aligned for these sizes)
  char* ws = (char*)d_ws;
  const size_t szX = (size_t)N_uiu * D * sizeof(float);
  const size_t szH = (size_t)N_ii  * D * sizeof(float);
  float* bufA = (float*)(ws);
  float* bufB = (float*)(ws + szX);
  float* hii  = (float*)(ws + 2 * szX);
  float* deg  = (float*)(ws + 2 * szX + szH);
  float* nrm  = (float*)(ws + 2 * szX + szH + (size_t)N_uiu * sizeof(float));

  // ================= Phase 1: item-item graph =================
  fillf_kernel<<<cdiv(N_ii, TB), TB, 0, stream>>>(deg, N_ii);
  deg_kernel<<<cdiv(E_ii, TB), TB, 0, stream>>>(dst_ii, w_ii, deg, E_ii);
  norm_kernel<<<cdiv(E_ii, TB), TB, 0, stream>>>(src_ii, dst_ii, w_ii, deg, nrm, E_ii);

  const long n4_ii = (long)N_ii * 16;  // float4 count
  copy_scale_kernel<<<cdiv(n4_ii, TB), TB, 0, stream>>>(
      (const float4*)emb_ii, (float4*)bufA, (float4*)hii, alpha, n4_ii);

  float* pa = bufA;
  float* pb = bufB;
  for (int l = 0; l < L; ++l) {
    fill4_kernel<<<cdiv(n4_ii, TB), TB, 0, stream>>>((float4*)pb, n4_ii);
    spmm_kernel<<<cdiv(E_ii, CHUNK), TB, 0, stream>>>(src_ii, dst_ii, nrm, pa, pb, E_ii);
    run_combine(hii, pb, alpha, N_ii, stream);
    float* t = pa; pa = pb; pb = t;
  }
  // hii now holds h_ii

  // ================= Phase 2: user-item graph =================
  pa = bufA; pb = bufB;
  const long n4_u = (long)N_uiu * 16;
  copy4_kernel<<<cdiv(n4_u, TB), TB, 0, stream>>>((const float4*)emb_uiu, (float4*)pa, n4_u);
  item_scatter_kernel<<<cdiv((long)n_items * 16, TB), TB, 0, stream>>>(
      itemidx, (const float4*)hii, (float4*)pa, n_items);
  scale4_kernel<<<cdiv(n4_u, TB), TB, 0, stream>>>((const float4*)pa, (float4*)out, alpha, n4_u);

  fillf_kernel<<<cdiv(N_uiu, TB), TB, 0, stream>>>(deg, N_uiu);
  deg_kernel<<<cdiv(E_uiu, TB), TB, 0, stream>>>(dst_u, w_uiu, deg, E_uiu);
  norm_kernel<<<cdiv(E_uiu, TB), TB, 0, stream>>>(src_u, dst_u, w_uiu, deg, nrm, E_uiu);

  for (int l = 0; l < L; ++l) {
    fill4_kernel<<<cdiv(n4_u, TB), TB, 0, stream>>>((float4*)pb, n4_u);
    spmm_kernel<<<cdiv(E_uiu, CHUNK), TB, 0, stream>>>(src_u, dst_u, nrm, pa, pb, E_uiu);
    run_combine(out, pb, alpha, N_uiu, stream);
    float* t = pa; pa = pb; pb = t;
  }
}